// TemporalAttentionConv_47141561041027
// MI455X (gfx1250) — compile-verified
//
#include <hip/hip_runtime.h>

// ---- problem constants (from reference) ----
#define NN   64
#define CC   64
#define TT   128
#define VV   25
#define RR   8
#define OO   128
#define KSZ  5
#define PADT 2
#define TV   (TT*VV)        // 3200
#define NTV  (NN*TV)        // 204800
#define TILES (NTV/16)      // 12800
#define CNTF 204800.0f      // N*T*V, BN population count
#define EPSF 1e-5f
#define NMOM (RR + RR*(RR+1)/2)   // 8 means + 36 upper-tri second moments = 44

typedef __attribute__((ext_vector_type(2))) float v2f;
typedef __attribute__((ext_vector_type(8))) float v8f;

// wave32 butterfly reduction
__device__ __forceinline__ float waveRed(float v) {
  #pragma unroll
  for (int m = 16; m >= 1; m >>= 1) v += __shfl_xor(v, m, 32);
  return v;
}

// =====================================================================
// Stage 1: one WMMA chain computes v0 = Wv@x + bv AND r0 = Wr@x + br.
// A (16x64) = [Wv rows 0-7 ; Wr rows 8-15], B = x[C=64, 16 positions].
// Also emits xm = mean_C(x) (harvested from B loads) and fuses the
// r0 per-channel BN batch stats (sum / sumsq) via wave reduce + LDS.
// f32 WMMA layouts (ISA 7.12.2):
//   A 16x4 : v0: lanes0-15 K=0 | lanes16-31 K=2 ; v1: K=1 | K=3  (M = lane%16)
//   B 4x16 : v0: lanes0-15 K=0 | lanes16-31 K=2 ; v1: K=1 | K=3  (N = lane%16)
//   C 16x16: vgpr j: lanes0-15 M=j | lanes16-31 M=j+8, N = lane%16
// =====================================================================
__global__ __launch_bounds__(256)
void tac_stage1(const float* __restrict__ x,  const float* __restrict__ Wv,
                const float* __restrict__ bv, const float* __restrict__ Wr,
                const float* __restrict__ br,
                float* __restrict__ v0, float* __restrict__ r0,
                float* __restrict__ xm, float* __restrict__ statsR)
{
  __shared__ float shS[RR], shS2[RR];
  if (threadIdx.x < RR) { shS[threadIdx.x] = 0.0f; shS2[threadIdx.x] = 0.0f; }
  __syncthreads();

  const int lane = threadIdx.x & 31;
  const int wave = threadIdx.x >> 5;
  const int tile = blockIdx.x * 8 + wave;          // exact cover: TILES/8 blocks
  const int n    = tile / (TV/16);
  const int p0   = (tile % (TV/16)) * 16;
  const int half = lane >> 4;                      // 0: lanes 0-15, 1: lanes 16-31
  const int l    = lane & 15;

  const float* Arow = (l < RR) ? (Wv + l * CC) : (Wr + (l - RR) * CC);
  const float* xb   = x + ((size_t)n * CC) * TV + p0;   // x[n, c, p0 + col]

  v8f acc = {};
  float bsum = 0.0f;
  #pragma unroll
  for (int c = 0; c < CC; c += 4) {
    v2f a, b;
    a.x = Arow[c + 2*half + 0];
    a.y = Arow[c + 2*half + 1];
    const float* bp = xb + (size_t)(c + 2*half) * TV;
    b.x = bp[l];            // row K = c+2*half      at column l
    b.y = bp[TV + l];       // row K = c+2*half + 1  at column l
    bsum += b.x + b.y;
    acc = __builtin_amdgcn_wmma_f32_16x16x4_f32(false, a, false, b,
                                                (short)0, acc, false, false);
  }

  // channel mean: combine the two lane-halves holding the same position
  const float tot = bsum + __shfl_xor(bsum, 16, 32);
  if (half == 0) xm[(size_t)n * TV + p0 + l] = tot * (1.0f / CC);

  float*       outT = half ? r0 : v0;   // rows 0-7 = v0 ; rows 8-15 = r0
  const float* bias = half ? br : bv;
  float outv[RR];
  #pragma unroll
  for (int j = 0; j < RR; ++j) {
    outv[j] = acc[j] + bias[j];
    outT[((size_t)n * RR + j) * TV + p0 + l] = outv[j];
  }

  // fused r0 batch stats (half==1 lanes hold the r0 values)
  #pragma unroll
  for (int j = 0; j < RR; ++j) {
    const float rv = half ? outv[j] : 0.0f;
    const float s  = waveRed(rv);
    const float s2 = waveRed(rv * rv);
    if (lane == 0) { atomicAdd(&shS[j], s); atomicAdd(&shS2[j], s2); }
  }
  __syncthreads();
  if (threadIdx.x < RR) {
    atomicAdd(&statsR[2 * threadIdx.x],     shS[threadIdx.x]);
    atomicAdd(&statsR[2 * threadIdx.x + 1], shS2[threadIdx.x]);
  }
}

// =====================================================================
// Attention: ws[u] = sum_w tanh(Ww[w*5+u,:] . m[:,t,v] + bw), m = unfold(xm);
// att[r] = sum_u ws[u] * v0[r, t+u-2, v]  (zero-padded in T).
// Fuses att per-channel BN batch stats.
// =====================================================================
__global__ __launch_bounds__(256)
void tac_attn(const float* __restrict__ xm, const float* __restrict__ Ww,
              const float* __restrict__ bw, const float* __restrict__ v0,
              float* __restrict__ att, float* __restrict__ statsA)
{
  __shared__ float shS[RR], shS2[RR];
  if (threadIdx.x < RR) { shS[threadIdx.x] = 0.0f; shS2[threadIdx.x] = 0.0f; }
  __syncthreads();

  const int p = blockIdx.x * 256 + threadIdx.x;      // exact cover of NTV
  const int v = p % VV;
  const int t = (p / VV) % TT;
  const int n = p / TV;
  const int lane = threadIdx.x & 31;

  float mtap[KSZ];
  #pragma unroll
  for (int k = 0; k < KSZ; ++k) {
    const int tt = t + k - PADT;
    mtap[k] = (tt >= 0 && tt < TT) ? xm[((size_t)n * TT + tt) * VV + v] : 0.0f;
  }

  float wsum[KSZ] = {0.f, 0.f, 0.f, 0.f, 0.f};
  #pragma unroll
  for (int w = 0; w < KSZ; ++w)
    #pragma unroll
    for (int u = 0; u < KSZ; ++u) {
      const int o = w * KSZ + u;
      float a = bw[o];
      #pragma unroll
      for (int k = 0; k < KSZ; ++k) a += Ww[o * KSZ + k] * mtap[k];
      wsum[u] += tanhf(a);
    }

  #pragma unroll
  for (int r = 0; r < RR; ++r) {
    float a = 0.0f;
    #pragma unroll
    for (int u = 0; u < KSZ; ++u) {
      const int tt = t + u - PADT;
      if (tt >= 0 && tt < TT)
        a += wsum[u] * v0[(((size_t)n * RR + r) * TT + tt) * VV + v];
    }
    att[(((size_t)n * RR + r) * TT + t) * VV + v] = a;
    const float s  = waveRed(a);
    const float s2 = waveRed(a * a);
    if (lane == 0) { atomicAdd(&shS[r], s); atomicAdd(&shS2[r], s2); }
  }
  __syncthreads();
  if (threadIdx.x < RR) {
    atomicAdd(&statsA[2 * threadIdx.x],     shS[threadIdx.x]);
    atomicAdd(&statsA[2 * threadIdx.x + 1], shS2[threadIdx.x]);
  }
}

// =====================================================================
// h moment pass: h = LeakyReLU(BN(att) + BN(r0)); accumulate the 8 means
// and 36 upper-triangular second moments of h over (N,T,V). These fully
// determine the final BN's batch stats (y = Wo h + bo is linear in h),
// so d_out never needs a stats pass.
// =====================================================================
__global__ __launch_bounds__(256)
void tac_hmom(const float* __restrict__ att, const float* __restrict__ r0,
              const float* __restrict__ statsR, const float* __restrict__ statsA,
              const float* __restrict__ g_bn, const float* __restrict__ b_bn,
              const float* __restrict__ g_r,  const float* __restrict__ b_r,
              float* __restrict__ hmom)
{
  __shared__ float sh[NMOM];
  if (threadIdx.x < NMOM) sh[threadIdx.x] = 0.0f;
  __syncthreads();

  const int p = blockIdx.x * 256 + threadIdx.x;      // exact cover of NTV
  const int n = p / TV, q = p - n * TV;
  const int lane = threadIdx.x & 31;

  float h[RR];
  #pragma unroll
  for (int r = 0; r < RR; ++r) {
    const size_t idx = ((size_t)n * RR + r) * TV + q;
    const float muA = statsA[2*r]   * (1.0f / CNTF);
    const float vaA = statsA[2*r+1] * (1.0f / CNTF) - muA * muA;
    const float muR = statsR[2*r]   * (1.0f / CNTF);
    const float vaR = statsR[2*r+1] * (1.0f / CNTF) - muR * muR;
    const float hv = (att[idx] - muA) * rsqrtf(vaA + EPSF) * g_bn[r] + b_bn[r]
                   + (r0[idx]  - muR) * rsqrtf(vaR + EPSF) * g_r[r]  + b_r[r];
    h[r] = hv >= 0.0f ? hv : 0.1f * hv;
  }

  int k = 0;
  #pragma unroll
  for (int i = 0; i < RR; ++i) {
    const float s = waveRed(h[i]);
    if (lane == 0) atomicAdd(&sh[k], s);
    ++k;
  }
  #pragma unroll
  for (int i = 0; i < RR; ++i)
    #pragma unroll
    for (int j = i; j < RR; ++j) {
      const float s = waveRed(h[i] * h[j]);
      if (lane == 0) atomicAdd(&sh[k], s);
      ++k;
    }
  __syncthreads();
  if (threadIdx.x < NMOM) atomicAdd(&hmom[threadIdx.x], sh[threadIdx.x]);
}

// =====================================================================
// Convert h moments -> final BN scale/shift per output channel:
//   mu_o  = Wo_o . m̄ + bo_o
//   var_o = Wo_oᵀ Cov(h) Wo_o
//   out   = y*scale + shift with scale = g*rsqrt(var+eps), shift = b - mu*scale
// =====================================================================
__global__ __launch_bounds__(OO)
void tac_yscale(const float* __restrict__ hmom, const float* __restrict__ Wo,
                const float* __restrict__ bo,   const float* __restrict__ g_o,
                const float* __restrict__ b_o,
                float* __restrict__ scaleY, float* __restrict__ shiftY)
{
  const int o = threadIdx.x;          // 128 threads, 1 block
  float w[RR], mbar[RR];
  #pragma unroll
  for (int i = 0; i < RR; ++i) {
    w[i]    = Wo[(size_t)o * RR + i];
    mbar[i] = hmom[i] * (1.0f / CNTF);
  }
  float mu = bo[o];
  #pragma unroll
  for (int i = 0; i < RR; ++i) mu += w[i] * mbar[i];

  float var = 0.0f;
  int k = RR;
  #pragma unroll
  for (int i = 0; i < RR; ++i)
    #pragma unroll
    for (int j = i; j < RR; ++j) {
      const float cov = hmom[k] * (1.0f / CNTF) - mbar[i] * mbar[j];
      var += (i == j ? 1.0f : 2.0f) * w[i] * w[j] * cov;
      ++k;
    }
  const float sc = g_o[o] * rsqrtf(var + EPSF);
  scaleY[o] = sc;
  shiftY[o] = b_o[o] - mu * sc;
}

// =====================================================================
// Stage 2: h = BN(att)+BN(r0), LeakyReLU(0.1); y = Wo @ h + bo; final BN
// applied inline via precomputed scale/shift -> d_out written exactly once.
// =====================================================================
__global__ __launch_bounds__(256)
void tac_stage2(const float* __restrict__ att, const float* __restrict__ r0,
                const float* __restrict__ statsR, const float* __restrict__ statsA,
                const float* __restrict__ g_bn, const float* __restrict__ b_bn,
                const float* __restrict__ g_r,  const float* __restrict__ b_r,
                const float* __restrict__ Wo,   const float* __restrict__ bo,
                const float* __restrict__ scaleY, const float* __restrict__ shiftY,
                float* __restrict__ out)
{
  const int lane = threadIdx.x & 31;
  const int wave = threadIdx.x >> 5;
  const int tile = blockIdx.x * 8 + wave;
  const int n    = tile / (TV/16);
  const int p0   = (tile % (TV/16)) * 16;
  const int half = lane >> 4;
  const int l    = lane & 15;

  auto hval = [&](int r) -> float {
    const size_t idx = ((size_t)n * RR + r) * TV + p0 + l;
    const float muA = statsA[2*r]   * (1.0f / CNTF);
    const float vaA = statsA[2*r+1] * (1.0f / CNTF) - muA * muA;
    const float muR = statsR[2*r]   * (1.0f / CNTF);
    const float vaR = statsR[2*r+1] * (1.0f / CNTF) - muR * muR;
    const float hv = (att[idx] - muA) * rsqrtf(vaA + EPSF) * g_bn[r] + b_bn[r]
                   + (r0[idx]  - muR) * rsqrtf(vaR + EPSF) * g_r[r]  + b_r[r];
    return hv >= 0.0f ? hv : 0.1f * hv;   // LeakyReLU(0.1)
  };

  // B operand (h, 8x16): step0 rows K=0..3, step1 rows K=4..7
  v2f b0, b1;
  b0.x = hval(2*half);     b0.y = hval(2*half + 1);
  b1.x = hval(4 + 2*half); b1.y = hval(4 + 2*half + 1);

  #pragma unroll
  for (int mt = 0; mt < 8; ++mt) {
    const float* Arow = Wo + (size_t)(mt * 16 + l) * RR;
    v2f a0, a1;
    a0.x = Arow[2*half];     a0.y = Arow[2*half + 1];
    a1.x = Arow[4 + 2*half]; a1.y = Arow[4 + 2*half + 1];
    v8f c = {};
    c = __builtin_amdgcn_wmma_f32_16x16x4_f32(false, a0, false, b0, (short)0, c, false, false);
    c = __builtin_amdgcn_wmma_f32_16x16x4_f32(false, a1, false, b1, (short)0, c, false, false);
    #pragma unroll
    for (int j = 0; j < 8; ++j) {
      const int o = mt * 16 + j + 8 * half;   // D vgpr j: M=j (half0) / j+8 (half1)
      const float ypre = c[j] + bo[o];
      out[((size_t)n * OO + o) * TV + p0 + l] = ypre * scaleY[o] + shiftY[o];
    }
  }
}

// =====================================================================
extern "C" void kernel_launch(void* const* d_in, const int* in_sizes, int n_in,
                              void* d_out, int out_size, void* d_ws, size_t ws_size,
                              hipStream_t stream)
{
  const float* x    = (const float*)d_in[0];
  const float* Wv   = (const float*)d_in[1];
  const float* bv   = (const float*)d_in[2];
  const float* Ww   = (const float*)d_in[3];
  const float* bw   = (const float*)d_in[4];
  const float* g_bn = (const float*)d_in[5];
  const float* b_bn = (const float*)d_in[6];
  const float* Wr   = (const float*)d_in[7];
  const float* br   = (const float*)d_in[8];
  const float* g_r  = (const float*)d_in[9];
  const float* b_r  = (const float*)d_in[10];
  const float* Wo   = (const float*)d_in[11];
  const float* bo   = (const float*)d_in[12];
  const float* g_o  = (const float*)d_in[13];
  const float* b_o  = (const float*)d_in[14];

  float* ws    = (float*)d_ws;
  float* v0    = ws;                               // N*R*T*V
  float* r0    = v0  + (size_t)NN * RR * TV;       // N*R*T*V
  float* xm    = r0  + (size_t)NN * RR * TV;       // N*T*V
  float* att   = xm  + (size_t)NN * TV;            // N*R*T*V
  float* stats = att + (size_t)NN * RR * TV;
  float* statsR = stats;                 // 16
  float* statsA = stats + 2 * RR;        // 16
  float* hmom   = stats + 4 * RR;        // 44
  float* scaleY = hmom + NMOM;           // 128
  float* shiftY = scaleY + OO;           // 128

  float* y = (float*)d_out;

  // zero the accumulated stats (statsR + statsA + hmom)
  hipMemsetAsync(stats, 0, (4 * RR + NMOM) * sizeof(float), stream);

  tac_stage1<<<TILES/8, 256, 0, stream>>>(x, Wv, bv, Wr, br, v0, r0, xm, statsR);
  tac_attn  <<<NTV/256, 256, 0, stream>>>(xm, Ww, bw, v0, att, statsA);
  tac_hmom  <<<NTV/256, 256, 0, stream>>>(att, r0, statsR, statsA,
                                          g_bn, b_bn, g_r, b_r, hmom);
  tac_yscale<<<1, OO, 0, stream>>>(hmom, Wo, bo, g_o, b_o, scaleY, shiftY);
  tac_stage2<<<TILES/8, 256, 0, stream>>>(att, r0, statsR, statsA,
                                          g_bn, b_bn, g_r, b_r, Wo, bo,
                                          scaleY, shiftY, y);
}